// DGCNN_71837622993125
// MI455X (gfx1250) — compile-verified
//
#include <hip/hip_runtime.h>
#include <hip/hip_bf16.h>
#include <math.h>

#define NN      50000
#define NE      800000
#define NEL     (NE + NN)   // edges + self loops
#define NG      128
#define FEAT_D  128
#define HID_D   64
#define KSEL    30
#define LATENT  193   // HID*3 + 1

typedef float v2f __attribute__((ext_vector_type(2)));
typedef float v8f __attribute__((ext_vector_type(8)));

static inline int cdiv(long long a, long long b) { return (int)((a + b - 1) / b); }

__device__ inline void atomAddF(float* p, float v) {
  __hip_atomic_fetch_add(p, v, __ATOMIC_RELAXED, __HIP_MEMORY_SCOPE_AGENT);
}

// ---------------- fills ----------------
__global__ void k_fillf(float* __restrict__ p, int n) {
  int i = blockIdx.x * blockDim.x + threadIdx.x;
  if (i < n) p[i] = 0.0f;
}
__global__ void k_filli(int* __restrict__ p, int n) {
  int i = blockIdx.x * blockDim.x + threadIdx.x;
  if (i < n) p[i] = 0;
}

// ---------------- degree / norm ----------------
__global__ void k_deg(const int* __restrict__ ei, float* __restrict__ deg) {
  int e = blockIdx.x * blockDim.x + threadIdx.x;
  if (e < NE) atomAddF(&deg[ei[NE + e]], 1.0f);
}
__global__ void k_dinv(const float* __restrict__ deg, float* __restrict__ dinv) {
  int i = blockIdx.x * blockDim.x + threadIdx.x;
  if (i < NN) dinv[i] = rsqrtf(deg[i] + 1.0f);  // +1 = self loop; always > 0
}
// Flatten edges + self-loops into branch-free (src, dst, norm) arrays.
__global__ void k_edgeprep(const int* __restrict__ ei, const float* __restrict__ dinv,
                           int* __restrict__ esrc, int* __restrict__ edst,
                           float* __restrict__ normE) {
  int e = blockIdx.x * blockDim.x + threadIdx.x;
  if (e >= NEL) return;
  int s, d;
  if (e < NE) { s = ei[e]; d = ei[NE + e]; }
  else        { s = d = e - NE; }
  esrc[e] = s;
  edst[e] = d;
  normE[e] = dinv[s] * dinv[d];
}

// ---------------- fp32 WMMA GEMM: C[MxN] = A[MxK] @ B[KxN] ----------------
// One wave computes a 16x64 strip of C (4 accumulators) so each A fragment
// feeds 4 v_wmma ops. Requires M%16==0, N%64==0, K%4==0.
// Fragment layouts per CDNA5 ISA 7.12.2:
//   A 16x4 : lanes 0-15 -> {K0,K1}, lanes 16-31 -> {K2,K3}, row M = lane&15
//   B 4x16 : VGPR0/1 hold rows K=(half*2), K=(half*2+1), col N = lane&15
//   C 16x16: VGPR v -> row v (lanes 0-15) / row v+8 (lanes 16-31), col = lane&15
__global__ __launch_bounds__(32) void k_gemm_wmma(const float* __restrict__ A,
                                                  const float* __restrict__ B,
                                                  float* __restrict__ C,
                                                  int M, int N, int K) {
  const int tm   = blockIdx.x;       // 16-row strip
  const int tn   = blockIdx.y;       // 64-col strip
  const int lane = threadIdx.x;
  const int half = lane >> 4;
  const int lm   = lane & 15;
  const int arow = tm * 16 + lm;
  const int cbase = tn * 64 + lm;

  v8f c0 = {}, c1 = {}, c2 = {}, c3 = {};
  for (int k0 = 0; k0 < K; k0 += 4) {
    v2f a;
    a.x = A[(size_t)arow * K + k0 + 2 * half];
    a.y = A[(size_t)arow * K + k0 + 2 * half + 1];
    const float* Brow0 = B + (size_t)(k0 + 2 * half) * N + cbase;
    const float* Brow1 = B + (size_t)(k0 + 2 * half + 1) * N + cbase;
    v2f b;
    b.x = Brow0[0];  b.y = Brow1[0];
    c0 = __builtin_amdgcn_wmma_f32_16x16x4_f32(false, a, false, b, (short)0, c0, false, false);
    b.x = Brow0[16]; b.y = Brow1[16];
    c1 = __builtin_amdgcn_wmma_f32_16x16x4_f32(false, a, false, b, (short)0, c1, false, false);
    b.x = Brow0[32]; b.y = Brow1[32];
    c2 = __builtin_amdgcn_wmma_f32_16x16x4_f32(false, a, false, b, (short)0, c2, false, false);
    b.x = Brow0[48]; b.y = Brow1[48];
    c3 = __builtin_amdgcn_wmma_f32_16x16x4_f32(false, a, false, b, (short)0, c3, false, false);
  }
  const int crow = tm * 16 + half * 8;
#pragma unroll
  for (int v = 0; v < 8; ++v) {
    float* Cr = C + (size_t)(crow + v) * N + cbase;
    Cr[0]  = c0[v];
    Cr[16] = c1[v];
    Cr[32] = c2[v];
    Cr[48] = c3[v];
  }
}

// ---------------- GCN scatter: agg[dst] += h[src] * norm -------------------
// 16 threads per edge, each handles 4 features via float4 load + 4 atomics.
__global__ void k_scatter64(const float* __restrict__ h, const int* __restrict__ esrc,
                            const int* __restrict__ edst, const float* __restrict__ normE,
                            float* __restrict__ agg) {
  long long idx = (long long)blockIdx.x * blockDim.x + threadIdx.x;
  if (idx >= (long long)NEL * 16) return;
  int q = (int)(idx & 15);          // feature quad
  int e = (int)(idx >> 4);
  int s = esrc[e];
  int d = edst[e];
  float w = normE[e];
  const float4 hv = *(const float4*)(h + (size_t)s * HID_D + q * 4);
  float* ag = agg + (size_t)d * HID_D + q * 4;
  atomAddF(ag + 0, hv.x * w);
  atomAddF(ag + 1, hv.y * w);
  atomAddF(ag + 2, hv.z * w);
  atomAddF(ag + 3, hv.w * w);
}

__global__ void k_scatter1(const float* __restrict__ hv, const int* __restrict__ esrc,
                           const int* __restrict__ edst, const float* __restrict__ normE,
                           float* __restrict__ agg) {
  int e = blockIdx.x * blockDim.x + threadIdx.x;
  if (e >= NEL) return;
  atomAddF(&agg[edst[e]], hv[esrc[e]] * normE[e]);
}

__global__ void k_bias_tanh(const float* __restrict__ agg, const float* __restrict__ b,
                            float* __restrict__ h) {
  int i = blockIdx.x * blockDim.x + threadIdx.x;
  if (i >= NN * HID_D) return;
  h[i] = tanhf(agg[i] + b[i & (HID_D - 1)]);
}

__global__ void k_matvec64(const float* __restrict__ h3, const float* __restrict__ w3,
                           float* __restrict__ hv) {
  int i = blockIdx.x * blockDim.x + threadIdx.x;
  if (i >= NN) return;
  float acc = 0.0f;
#pragma unroll
  for (int k = 0; k < HID_D; ++k) acc += h3[(size_t)i * HID_D + k] * w3[k];
  hv[i] = acc;
}

__global__ void k_tanh1(const float* __restrict__ agg, const float* __restrict__ b3,
                        float* __restrict__ h4) {
  int i = blockIdx.x * blockDim.x + threadIdx.x;
  if (i < NN) h4[i] = tanhf(agg[i] + b3[0]);
}

// ---------------- sort-pool ----------------
__global__ void k_counts(const int* __restrict__ batch, int* __restrict__ counts) {
  int i = blockIdx.x * blockDim.x + threadIdx.x;
  if (i < NN) atomicAdd(&counts[batch[i]], 1);
}
__global__ void k_starts(const int* __restrict__ counts, int* __restrict__ starts) {
  if (blockIdx.x == 0 && threadIdx.x == 0) {
    int acc = 0;
    for (int g = 0; g < NG; ++g) { starts[g] = acc; acc += counts[g]; }
  }
}

// Per-graph top-K by h4 descending, stable (smaller node index wins ties).
__global__ __launch_bounds__(128) void k_topk(const float* __restrict__ h4,
                                              const int* __restrict__ starts,
                                              const int* __restrict__ counts,
                                              int* __restrict__ sel) {
  const int g = blockIdx.x;
  const int tid = threadIdx.x;
  const int start = starts[g];
  const int cnt = counts[g];
  __shared__ int picked[KSEL];
  __shared__ float sval[128];
  __shared__ int sidx[128];
  for (int r = 0; r < KSEL; ++r) {
    float bestv = -INFINITY;
    int besti = 0x7fffffff;
    for (int i = tid; i < cnt; i += 128) {
      int node = start + i;
      bool used = false;
      for (int p = 0; p < r; ++p) used |= (picked[p] == node);
      if (used) continue;
      float v = h4[node];
      if (v > bestv || (v == bestv && node < besti)) { bestv = v; besti = node; }
    }
    sval[tid] = bestv; sidx[tid] = besti;
    __syncthreads();
    for (int s = 64; s > 0; s >>= 1) {
      if (tid < s) {
        float ov = sval[tid + s]; int oi = sidx[tid + s];
        if (ov > sval[tid] || (ov == sval[tid] && oi < sidx[tid])) {
          sval[tid] = ov; sidx[tid] = oi;
        }
      }
      __syncthreads();
    }
    if (tid == 0) {
      int chosen = (r < cnt) ? sidx[0] : -1;
      picked[r] = chosen;
      sel[g * KSEL + r] = chosen;
    }
    __syncthreads();
  }
}

// pooled[g][r][j] = hcat[sel[g][r]][j], zero if no node.
__global__ void k_pool(const int* __restrict__ sel,
                       const float* __restrict__ h1, const float* __restrict__ h2,
                       const float* __restrict__ h3, const float* __restrict__ h4,
                       float* __restrict__ pooled) {
  int idx = blockIdx.x * blockDim.x + threadIdx.x;
  if (idx >= NG * KSEL * LATENT) return;
  int j = idx % LATENT;
  int gr = idx / LATENT;
  int node = sel[gr];
  float v = 0.0f;
  if (node >= 0) {
    if (j < 64)       v = h1[(size_t)node * HID_D + j];
    else if (j < 128) v = h2[(size_t)node * HID_D + j - 64];
    else if (j < 192) v = h3[(size_t)node * HID_D + j - 128];
    else              v = h4[node];
  }
  pooled[idx] = v;
}

// ---------------- CNN tail ----------------
// conv1: kernel 193 stride 193 over (NG,1,30*193) -> (NG,16,30), ReLU
__global__ void k_conv1(const float* __restrict__ pooled, const float* __restrict__ w,
                        const float* __restrict__ b, float* __restrict__ z1) {
  int idx = blockIdx.x * blockDim.x + threadIdx.x;
  if (idx >= NG * 16 * KSEL) return;
  int t  = idx % KSEL;
  int co = (idx / KSEL) % 16;
  int g  = idx / (16 * KSEL);
  const float* p  = pooled + (size_t)(g * KSEL + t) * LATENT;
  const float* wr = w + (size_t)co * LATENT;
  float acc = b[co];
  for (int j = 0; j < LATENT; ++j) acc += p[j] * wr[j];
  z1[idx] = fmaxf(acc, 0.0f);
}

// maxpool size2 stride2: (NG,16,30) -> (NG,16,15)
__global__ void k_maxpool(const float* __restrict__ z1, float* __restrict__ z2) {
  int idx = blockIdx.x * blockDim.x + threadIdx.x;
  if (idx >= NG * 16 * 15) return;
  int t = idx % 15;
  int base = (idx / 15) * KSEL;
  z2[idx] = fmaxf(z1[base + 2 * t], z1[base + 2 * t + 1]);
}

// conv2: (NG,16,15) * (32,16,5) stride1 -> (NG,32,11), ReLU, flattened [NG][352]
__global__ void k_conv2(const float* __restrict__ z2, const float* __restrict__ w,
                        const float* __restrict__ b, float* __restrict__ z3) {
  int idx = blockIdx.x * blockDim.x + threadIdx.x;
  if (idx >= NG * 32 * 11) return;
  int t  = idx % 11;
  int co = (idx / 11) % 32;
  int g  = idx / (32 * 11);
  float acc = b[co];
  for (int ci = 0; ci < 16; ++ci) {
#pragma unroll
    for (int j = 0; j < 5; ++j)
      acc += z2[g * 240 + ci * 15 + t + j] * w[co * 80 + ci * 5 + j];
  }
  z3[(size_t)g * 352 + co * 11 + t] = fmaxf(acc, 0.0f);
}

__global__ void k_bias_relu(const float* __restrict__ zg, const float* __restrict__ b,
                            float* __restrict__ z, int n, int cols) {
  int i = blockIdx.x * blockDim.x + threadIdx.x;
  if (i < n) z[i] = fmaxf(zg[i] + b[i % cols], 0.0f);
}

__global__ void k_lin2(const float* __restrict__ z4, const float* __restrict__ w,
                       const float* __restrict__ b, float* __restrict__ out) {
  int idx = blockIdx.x * blockDim.x + threadIdx.x;
  if (idx >= NG * 2) return;
  int o = idx % 2;
  int g = idx / 2;
  float acc = b[o];
  for (int k = 0; k < 256; ++k) acc += z4[(size_t)g * 256 + k] * w[k * 2 + o];
  out[idx] = acc;
}

// ---------------- host ----------------
extern "C" void kernel_launch(void* const* d_in, const int* in_sizes, int n_in,
                              void* d_out, int out_size, void* d_ws, size_t ws_size,
                              hipStream_t stream) {
  const float* x       = (const float*)d_in[0];
  const int*   ei      = (const int*)d_in[1];   // (2, NE): [0..NE)=src, [NE..2NE)=dst
  const int*   batch   = (const int*)d_in[2];
  const float* w0      = (const float*)d_in[3];
  const float* b0      = (const float*)d_in[4];
  const float* w1      = (const float*)d_in[5];
  const float* b1      = (const float*)d_in[6];
  const float* w2      = (const float*)d_in[7];
  const float* b2      = (const float*)d_in[8];
  const float* w3      = (const float*)d_in[9];
  const float* b3      = (const float*)d_in[10];
  const float* c1w     = (const float*)d_in[11];
  const float* c1b     = (const float*)d_in[12];
  const float* c2w     = (const float*)d_in[13];
  const float* c2b     = (const float*)d_in[14];
  const float* l1w     = (const float*)d_in[15];
  const float* l1b     = (const float*)d_in[16];
  const float* l2w     = (const float*)d_in[17];
  const float* l2b     = (const float*)d_in[18];
  float* out = (float*)d_out;

  // workspace carve
  char* ws = (char*)d_ws;
  size_t off = 0;
  auto alloc = [&](size_t bytes) -> void* {
    void* p = ws + off;
    off = (off + bytes + 255) & ~(size_t)255;
    return p;
  };
  float* deg    = (float*)alloc((size_t)NN * 4);
  float* dinv   = (float*)alloc((size_t)NN * 4);
  int*   esrc   = (int*)alloc((size_t)NEL * 4);
  int*   edst   = (int*)alloc((size_t)NEL * 4);
  float* normE  = (float*)alloc((size_t)NEL * 4);
  float* hl     = (float*)alloc((size_t)NN * HID_D * 4);
  float* agg    = (float*)alloc((size_t)NN * HID_D * 4);
  float* h1     = (float*)alloc((size_t)NN * HID_D * 4);
  float* h2     = (float*)alloc((size_t)NN * HID_D * 4);
  float* h3     = (float*)alloc((size_t)NN * HID_D * 4);
  float* hv     = (float*)alloc((size_t)NN * 4);
  float* agg4   = (float*)alloc((size_t)NN * 4);
  float* h4     = (float*)alloc((size_t)NN * 4);
  int*   counts = (int*)alloc(NG * 4);
  int*   starts = (int*)alloc(NG * 4);
  int*   sel    = (int*)alloc(NG * KSEL * 4);
  float* pooled = (float*)alloc((size_t)NG * KSEL * LATENT * 4);
  float* z1     = (float*)alloc((size_t)NG * 16 * KSEL * 4);
  float* z2     = (float*)alloc((size_t)NG * 16 * 15 * 4);
  float* z3     = (float*)alloc((size_t)NG * 352 * 4);
  float* z4g    = (float*)alloc((size_t)NG * 256 * 4);
  float* z4     = (float*)alloc((size_t)NG * 256 * 4);

  const int T = 256;

  // degree + symmetric norm + flattened edge list
  k_fillf<<<cdiv(NN, T), T, 0, stream>>>(deg, NN);
  k_deg<<<cdiv(NE, T), T, 0, stream>>>(ei, deg);
  k_dinv<<<cdiv(NN, T), T, 0, stream>>>(deg, dinv);
  k_edgeprep<<<cdiv(NEL, T), T, 0, stream>>>(ei, dinv, esrc, edst, normE);

  // layer 0: x(50000x128) @ w0(128x64)
  k_gemm_wmma<<<dim3(NN / 16, HID_D / 64), 32, 0, stream>>>(x, w0, hl, NN, HID_D, FEAT_D);
  k_fillf<<<cdiv((long long)NN * HID_D, T), T, 0, stream>>>(agg, NN * HID_D);
  k_scatter64<<<cdiv((long long)NEL * 16, T), T, 0, stream>>>(hl, esrc, edst, normE, agg);
  k_bias_tanh<<<cdiv((long long)NN * HID_D, T), T, 0, stream>>>(agg, b0, h1);

  // layer 1
  k_gemm_wmma<<<dim3(NN / 16, HID_D / 64), 32, 0, stream>>>(h1, w1, hl, NN, HID_D, HID_D);
  k_fillf<<<cdiv((long long)NN * HID_D, T), T, 0, stream>>>(agg, NN * HID_D);
  k_scatter64<<<cdiv((long long)NEL * 16, T), T, 0, stream>>>(hl, esrc, edst, normE, agg);
  k_bias_tanh<<<cdiv((long long)NN * HID_D, T), T, 0, stream>>>(agg, b1, h2);

  // layer 2
  k_gemm_wmma<<<dim3(NN / 16, HID_D / 64), 32, 0, stream>>>(h2, w2, hl, NN, HID_D, HID_D);
  k_fillf<<<cdiv((long long)NN * HID_D, T), T, 0, stream>>>(agg, NN * HID_D);
  k_scatter64<<<cdiv((long long)NEL * 16, T), T, 0, stream>>>(hl, esrc, edst, normE, agg);
  k_bias_tanh<<<cdiv((long long)NN * HID_D, T), T, 0, stream>>>(agg, b2, h3);

  // layer 3 (HID -> 1)
  k_matvec64<<<cdiv(NN, T), T, 0, stream>>>(h3, w3, hv);
  k_fillf<<<cdiv(NN, T), T, 0, stream>>>(agg4, NN);
  k_scatter1<<<cdiv(NEL, T), T, 0, stream>>>(hv, esrc, edst, normE, agg4);
  k_tanh1<<<cdiv(NN, T), T, 0, stream>>>(agg4, b3, h4);

  // sort-pool
  k_filli<<<cdiv(NG, T), T, 0, stream>>>(counts, NG);
  k_counts<<<cdiv(NN, T), T, 0, stream>>>(batch, counts);
  k_starts<<<1, 1, 0, stream>>>(counts, starts);
  k_topk<<<NG, 128, 0, stream>>>(h4, starts, counts, sel);
  k_pool<<<cdiv((long long)NG * KSEL * LATENT, T), T, 0, stream>>>(sel, h1, h2, h3, h4, pooled);

  // CNN tail
  k_conv1<<<cdiv(NG * 16 * KSEL, T), T, 0, stream>>>(pooled, c1w, c1b, z1);
  k_maxpool<<<cdiv(NG * 16 * 15, T), T, 0, stream>>>(z1, z2);
  k_conv2<<<cdiv(NG * 32 * 11, T), T, 0, stream>>>(z2, c2w, c2b, z3);

  // dense: (128x352)@(352x256) via WMMA, then bias+ReLU, then final (256->2)
  k_gemm_wmma<<<dim3(NG / 16, 256 / 64), 32, 0, stream>>>(z3, l1w, z4g, NG, 256, 352);
  k_bias_relu<<<cdiv(NG * 256, T), T, 0, stream>>>(z4g, l1b, z4, NG * 256, 256);
  k_lin2<<<1, 256, 0, stream>>>(z4, l2w, l2b, out);
}